// SpatialMultiOrderAdaptiveWeight_44014824849688
// MI455X (gfx1250) — compile-verified
//
#include <hip/hip_runtime.h>
#include <hip/hip_bf16.h>
#include <stdint.h>

// SpatialMultiOrderAdaptiveWeight for x:[32,96,128,128] f32.
// Memory-bound separable stencil: async global->LDS tile copy (CDNA5 path),
// vertical then horizontal incremental min/max/sum for k=3,5,7.
// Double-buffered column buffers (1 barrier/row), boundary-specialized tiles,
// sigmoid replaced by a 4-entry select (wsum is an exact small integer).

#define H_DIM 128
#define W_DIM 128
#define TH    16                 // output rows per block
#define TROWS (TH + 6)           // tile rows incl. +-3 halo
#define CW    (W_DIM + 8)        // padded column-buffer width (need 134)
#define PINF  __builtin_inff()

#if __has_builtin(__builtin_amdgcn_global_load_async_to_lds_b128)
#define USE_ASYNC_LDS 1
#else
#define USE_ASYNC_LDS 0
#endif

typedef int v4i __attribute__((vector_size(16)));
typedef __attribute__((address_space(1))) v4i gv4i;   // global-AS vector
typedef __attribute__((address_space(3))) v4i lv4i;   // LDS-AS vector

__device__ __forceinline__ void wait_async_zero() {
#if __has_builtin(__builtin_amdgcn_s_wait_asynccnt)
    __builtin_amdgcn_s_wait_asynccnt(0);
#else
    asm volatile("s_wait_asynccnt 0x0" ::: "memory");
#endif
}

// Process all TH output rows of a tile. CHECK=true only for the top/bottom
// tiles of a plane (vertical out-of-range substitution needed).
template <bool CHECK>
__device__ __forceinline__ void do_tile(const float* __restrict__ xs,
                                        float (*cb)[9][CW],
                                        float* __restrict__ dst,
                                        int r0, int xc) {
    for (int ty = 0; ty < TH; ++ty) {
        float (*b)[CW] = cb[ty & 1];
        const int yo = r0 + ty;

        // ---- vertical reductions over rows yo-3 .. yo+3 at column xc ----
        float vn[7], vx[7], vs[7];
        #pragma unroll
        for (int i = 0; i < 7; ++i) {
            const float r = xs[(ty + i) * W_DIM + xc];   // slot always in range
            if (CHECK) {
                const int y   = yo - 3 + i;
                const bool ok = ((unsigned)y < (unsigned)H_DIM);
                vn[i] = ok ? r :  PINF;
                vx[i] = ok ? r : -PINF;
                vs[i] = ok ? r :  0.0f;
            } else {
                vn[i] = r; vx[i] = r; vs[i] = r;
            }
        }
        const float n3 = fminf(fminf(vn[2], vn[3]), vn[4]);
        const float n5 = fminf(fminf(vn[1], n3), vn[5]);
        const float n7 = fminf(fminf(vn[0], n5), vn[6]);
        const float m3 = fmaxf(fmaxf(vx[2], vx[3]), vx[4]);
        const float m5 = fmaxf(fmaxf(vx[1], m3), vx[5]);
        const float m7 = fmaxf(fmaxf(vx[0], m5), vx[6]);
        const float s3 = vs[2] + vs[3] + vs[4];
        const float s5 = s3 + vs[1] + vs[5];
        const float s7 = s5 + vs[0] + vs[6];

        b[0][xc + 3] = n3;  b[1][xc + 3] = n5;  b[2][xc + 3] = n7;
        b[3][xc + 3] = m3;  b[4][xc + 3] = m5;  b[5][xc + 3] = m7;
        b[6][xc + 3] = s3;  b[7][xc + 3] = s5;  b[8][xc + 3] = s7;
        __syncthreads();   // single barrier/row (double buffering covers WAR)

        // ---- horizontal combine (window k at col xc -> b[.][xc+3-r .. xc+3+r]) ----
        float hn3 = fminf(fminf(b[0][xc + 2], b[0][xc + 3]), b[0][xc + 4]);
        float hn5 = b[1][xc + 1];
        #pragma unroll
        for (int j = 2; j <= 5; ++j) hn5 = fminf(hn5, b[1][xc + j]);
        float hn7 = b[2][xc + 0];
        #pragma unroll
        for (int j = 1; j <= 6; ++j) hn7 = fminf(hn7, b[2][xc + j]);

        float hm3 = fmaxf(fmaxf(b[3][xc + 2], b[3][xc + 3]), b[3][xc + 4]);
        float hm5 = b[4][xc + 1];
        #pragma unroll
        for (int j = 2; j <= 5; ++j) hm5 = fmaxf(hm5, b[4][xc + j]);
        float hm7 = b[5][xc + 0];
        #pragma unroll
        for (int j = 1; j <= 6; ++j) hm7 = fmaxf(hm7, b[5][xc + j]);

        float hs3 = b[6][xc + 2] + b[6][xc + 3] + b[6][xc + 4];
        float hs5 = b[7][xc + 1];
        #pragma unroll
        for (int j = 2; j <= 5; ++j) hs5 += b[7][xc + j];
        float hs7 = b[8][xc + 0];
        #pragma unroll
        for (int j = 1; j <= 6; ++j) hs7 += b[8][xc + j];

        const float a3 = hs3 * (1.0f / 9.0f);
        const float a5 = hs5 * (1.0f / 25.0f);
        const float a7 = hs7 * (1.0f / 49.0f);

        int cnt = 0;
        cnt += ((a3 > hn3) && (a3 < hm3)) ? 1 : 0;
        cnt += ((a5 > hn5) && (a5 < hm5)) ? 1 : 0;
        cnt += ((a7 > hn7) && (a7 < hm7)) ? 1 : 0;

        // sigmoid(cnt - 1.5) for cnt in {0,1,2,3}: correctly-rounded f32 values.
        const float w = (cnt == 0) ? 0.18242553f
                      : (cnt == 1) ? 0.37754068f
                      : (cnt == 2) ? 0.62245935f
                                   : 0.81757450f;

        const float xv = xs[(ty + 3) * W_DIM + xc];   // center row value
        dst[(long long)yo * W_DIM + xc] = xv * w;
        // no second barrier: next row writes the other cb buffer
    }
}

__global__ __launch_bounds__(W_DIM) void
SpatialMultiOrderAdaptiveWeight_44014824849688_kernel(const float* __restrict__ x,
                                                      float* __restrict__ out) {
    __shared__ __align__(16) float xs[TROWS * W_DIM];   // raw tile rows
    __shared__ __align__(16) float cb[2][9][CW];        // double-buffered columns
    // cb[b] rows: 0..2 = min3/5/7, 3..5 = max3/5/7, 6..8 = sum3/5/7

    const int tid = threadIdx.x;         // == output column
    const int r0  = blockIdx.x * TH;     // first output row of tile
    const long long planeOff = (long long)blockIdx.y * (H_DIM * W_DIM);
    const float* src = x + planeOff;
    float*       dst = out + planeOff;

    // ---- Stage 0: copy valid tile rows [rb, re) into LDS (contiguous region) ----
    const int rb   = (r0 - 3 < 0) ? 0 : (r0 - 3);
    const int re   = (r0 + TH + 3 > H_DIM) ? H_DIM : (r0 + TH + 3);
    const int lyb  = rb - (r0 - 3);                 // LDS start row
    const int nchunk = ((re - rb) * W_DIM) >> 2;    // 16B chunks (W=128 -> exact)
    {
        const float* g = src + (long long)rb * W_DIM;
        float*       l = xs + lyb * W_DIM;
#if USE_ASYNC_LDS
        for (int c = tid; c < nchunk; c += W_DIM) {
            __builtin_amdgcn_global_load_async_to_lds_b128(
                (gv4i*)(float*)(g + c * 4),
                (lv4i*)(l + c * 4),
                0, 0);
        }
        wait_async_zero();
#else
        for (int c = tid; c < nchunk; c += W_DIM) {
            const float4 v = *(const float4*)(g + c * 4);
            *(float4*)(l + c * 4) = v;
        }
#endif
    }

    // ---- Stage 0b: init halo columns of BOTH cb buffers with identities ----
    // Out-of-image columns are entirely padding, so their vertical reduction
    // equals the identity: +inf (min), -inf (max), 0 (sum).
    if (tid < 108) {
        const int bbuf = tid / 54;
        const int t    = tid % 54;
        const int a    = t / 6;                    // which cb row
        const int p    = t % 6;                    // which halo slot
        const int idx  = (p < 3) ? p : (W_DIM + p); // 0,1,2, 131,132,133
        cb[bbuf][a][idx] = (a < 3) ? PINF : ((a < 6) ? -PINF : 0.0f);
    }
    __syncthreads();

    // Interior tiles never sample rows outside [0, H): r0-3 >= 0 and r0+TH+3 <= H.
    if (r0 >= 3 && r0 + TH + 3 <= H_DIM) {
        do_tile<false>(xs, cb, dst, r0, tid);
    } else {
        do_tile<true>(xs, cb, dst, r0, tid);
    }
}

extern "C" void kernel_launch(void* const* d_in, const int* in_sizes, int n_in,
                              void* d_out, int out_size, void* d_ws, size_t ws_size,
                              hipStream_t stream) {
    (void)n_in; (void)d_ws; (void)ws_size; (void)out_size;
    const float* x = (const float*)d_in[0];
    float* out     = (float*)d_out;
    const int planes = in_sizes[0] / (H_DIM * W_DIM);   // B*C = 3072
    dim3 grid(H_DIM / TH, planes);                      // (8, 3072)
    SpatialMultiOrderAdaptiveWeight_44014824849688_kernel<<<grid, W_DIM, 0, stream>>>(x, out);
}